// MaskLayer_85907935854810
// MI455X (gfx1250) — compile-verified
//
#include <hip/hip_runtime.h>

// MaskLayer: [B=512, H=14, W=14, C=512] f32 -> same shape f32.
// Bandwidth-bound (0.41 GB @ 23.3 TB/s ~= 17.6 us floor). No GEMM content ->
// no WMMA. CDNA5 path: TDM async tensor load (Global -> LDS) so each
// (b, 64-channel) tile is read from HBM exactly once; both passes (argmax +
// apply) run out of LDS; NT stores for the write-once output stream.
//
// NOTE: the TDM builtin takes descriptor VALUES (no pointer arg), so LLVM
// cannot see that it writes LDS. Without countermeasures it proves tile[] is
// never stored and folds every LDS load to undef (observed in round 1:
// ds=0, valu=8). We block that with (a) an opaque never-taken ds_store and
// (b) a memory-clobber compiler barrier after the sync.

#define IMG   14
#define SP    (IMG * IMG)   // 196 spatial positions
#define CTOT  512           // channels
#define BTOT  512           // batch
#define CBLK  64            // channels per block (one thread per channel)

typedef unsigned int tdm_u32x4 __attribute__((ext_vector_type(4)));
typedef int          tdm_i32x8 __attribute__((ext_vector_type(8)));
typedef int          tdm_i32x4 __attribute__((ext_vector_type(4)));

__global__ __launch_bounds__(CBLK)
void MaskLayer_85907935854810_kernel(const float* __restrict__ in,
                                     float* __restrict__ out) {
  // 196 x 64 f32 tile = 49 KB LDS. Only LDS object -> LDS byte offset 0.
  __shared__ float tile[SP * CBLK];

  const int b  = blockIdx.x >> 3;              // sample
  const int c0 = (blockIdx.x & 7) * CBLK;      // channel-chunk base
  const int t  = threadIdx.x;                  // channel lane within chunk

  // Opaque zero: compiler must assume the guarded store may execute, so it
  // can no longer treat tile[] as never-written and fold LDS loads to undef.
  int opaque_zero = 0;
  asm volatile("" : "+s"(opaque_zero));
  if (opaque_zero) tile[opaque_zero] = 0.0f;   // never executes at runtime

  // ---------------- TDM: async DMA of the tile Global -> LDS --------------
  // D# layout per CDNA5 ISA.
  // Tensor as seen by this descriptor: 2D, dim0 = 512 ch (stride 512 elems),
  // dim1 = 196 rows. Tile: dim0 = 64 ch @ c0, dim1 = 196 rows (50,176 B).
  if (t == 0) {
    unsigned long long gaddr =
        (unsigned long long)(const void*)in +
        ((unsigned long long)b * (unsigned long long)(SP * CTOT) +
         (unsigned long long)c0) * 4ull;

    tdm_u32x4 g0;
    g0[0] = 1u;                                     // count=1, user descriptor
    g0[1] = 0u;                                     // lds_addr = 0 bytes
    g0[2] = (unsigned int)(gaddr & 0xFFFFFFFFull);  // global_addr[31:0]
    g0[3] = (unsigned int)((gaddr >> 32) & 0x1FFFFFFull) | (2u << 30); // [56:32] | type=2

    tdm_i32x8 g1;
    g1[0] = (2 << 16);                              // data_size=2 -> 4B; mask=0; no pad/iter
    g1[1] = (CTOT & 0xFFFF) << 16;                  // tensor_dim0[15:0] = 512
    g1[2] = ((SP & 0xFFFF) << 16) |                 // tensor_dim1[15:0] = 196
            ((CTOT >> 16) & 0xFFFF);                // tensor_dim0[31:16]
    g1[3] = (CBLK << 16) |                          // tile_dim0 = 64
            ((SP >> 16) & 0xFFFF);                  // tensor_dim1[31:16]
    g1[4] = SP;                                     // tile_dim1 = 196; tile_dim2 = 0
    g1[5] = CTOT;                                   // tensor_dim0_stride[31:0] = 512
    g1[6] = 0;                                      // stride0[47:32] | stride1[15:0]
    g1[7] = 0;                                      // stride1[47:16]

    tdm_i32x4 g2 = {0, 0, 0, 0};                    // 2D tensor: groups 2/3 unused
    tdm_i32x4 g3 = {0, 0, 0, 0};

#if defined(__clang_major__) && (__clang_major__ >= 23)
    tdm_i32x8 g4 = {0, 0, 0, 0, 0, 0, 0, 0};
    __builtin_amdgcn_tensor_load_to_lds(g0, g1, g2, g3, g4, 0);
#else
    __builtin_amdgcn_tensor_load_to_lds(g0, g1, g2, g3, 0);
#endif
    __builtin_amdgcn_s_wait_tensorcnt(0);           // only issuing wave sees TENSORcnt
  }
  __syncthreads();                                  // publish LDS tile to both waves
  asm volatile("" ::: "memory");                    // order LDS reads after the DMA

  // ---------------- Pass 1: row argmax + per-column maxes ------------------
  float colmax[IMG];
#pragma unroll
  for (int w = 0; w < IMG; ++w) colmax[w] = -__builtin_huge_valf();

  float bestRow = -__builtin_huge_valf();
  int   rIdx    = 0;
#pragma unroll
  for (int h = 0; h < IMG; ++h) {
    float rmax = -__builtin_huge_valf();
#pragma unroll
    for (int w = 0; w < IMG; ++w) {
      float x   = tile[(h * IMG + w) * CBLK + t];
      rmax      = fmaxf(rmax, x);
      colmax[w] = fmaxf(colmax[w], x);
    }
    if (rmax > bestRow) { bestRow = rmax; rIdx = h; }  // strict > == first-occurrence argmax
  }

  float bestCol = -__builtin_huge_valf();
  int   cIdx    = 0;
#pragma unroll
  for (int w = 0; w < IMG; ++w)
    if (colmax[w] > bestCol) { bestCol = colmax[w]; cIdx = w; }

  // ---------------- Pass 2: apply mask + ReLU, NT stream-out ---------------
  const float tau   = (float)(0.5 / (double)SP);    // 0.5 / 196
  const float slope = (float)(4.0 / 14.0);          // BETA / IMG

  float* outp = out + ((unsigned long long)b * (unsigned long long)(SP * CTOT) +
                       (unsigned long long)(c0 + t));
#pragma unroll
  for (int h = 0; h < IMG; ++h) {
    int ar = h - rIdx; if (ar < 0) ar = -ar;
#pragma unroll
    for (int w = 0; w < IMG; ++w) {
      int aw = w - cIdx; if (aw < 0) aw = -aw;
      float val = 1.0f - slope * (float)(ar + aw);
      val = fmaxf(val, -1.0f);
      float x = tile[(h * IMG + w) * CBLK + t];
      float r = (tau * val) * x;
      __builtin_nontemporal_store(fmaxf(r, 0.0f),
                                  outp + (unsigned long long)(h * IMG + w) * CTOT);
    }
  }
}

extern "C" void kernel_launch(void* const* d_in, const int* in_sizes, int n_in,
                              void* d_out, int out_size, void* d_ws, size_t ws_size,
                              hipStream_t stream) {
  (void)in_sizes; (void)n_in; (void)out_size; (void)d_ws; (void)ws_size;
  const float* in  = (const float*)d_in[0];
  float*       out = (float*)d_out;
  dim3 grid(BTOT * (CTOT / CBLK));   // 512 samples * 8 channel chunks = 4096
  dim3 block(CBLK);                  // 64 threads = 2 wave32, one channel each
  MaskLayer_85907935854810_kernel<<<grid, block, 0, stream>>>(in, out);
}